// PNAPCSAFT_26869315404576
// MI455X (gfx1250) — compile-verified
//
#include <hip/hip_runtime.h>
#include <math.h>

// ---------------- problem constants ----------------
#define NN 50000
#define EE 200000
#define HH 128
#define GG 1000
#define AVGDEG 1.1330197f   // sum(hist[d]*log(d+1))/num_nodes
#define EPS 1e-5f

typedef float v2f __attribute__((ext_vector_type(2)));
typedef float v8f __attribute__((ext_vector_type(8)));

// fp32 WMMA 16x16x4: D = A(16x4)*B(4x16) + C
__device__ __forceinline__ v8f wmma4(v2f a, v2f b, v8f c) {
  return __builtin_amdgcn_wmma_f32_16x16x4_f32(false, a, false, b, (short)0, c,
                                               false, false);
}

// order-preserving float <-> uint encoding for atomic min/max
__device__ __forceinline__ unsigned fenc(float f) {
  unsigned i = __float_as_uint(f);
  return (i & 0x80000000u) ? ~i : (i | 0x80000000u);
}
__device__ __forceinline__ float fdec(unsigned u) {
  unsigned i = (u & 0x80000000u) ? (u & 0x7FFFFFFFu) : ~u;
  return __uint_as_float(i);
}

// CDNA5 async global->LDS copy, per-lane 16B, tracked on ASYNCcnt.
// Generic pointer to LDS: low 32 bits are the LDS byte offset.
__device__ __forceinline__ void async_b128(void* lds, const void* g) {
  asm volatile("global_load_async_to_lds_b128 %0, %1, off"
               :
               : "v"((unsigned)(unsigned long long)lds),
                 "v"((unsigned long long)g)
               : "memory");
}
__device__ __forceinline__ void async_wait0() {
  asm volatile("s_wait_asynccnt 0x0" ::: "memory");
}

// ---------------- utility kernels ----------------
__global__ void k_fill(unsigned* p, unsigned v, long n) {
  long i = (long)blockIdx.x * blockDim.x + threadIdx.x;
  if (i < n) p[i] = v;
}

__global__ void k_count(const int* __restrict__ ei, int* cnt, int E) {
  int i = blockIdx.x * blockDim.x + threadIdx.x;
  if (i < E) atomicAdd(&cnt[ei[E + i]], 1);
}

__global__ void k_atom(const int* __restrict__ x, const float* __restrict__ emb,
                       float* __restrict__ h) {
  long i = (long)blockIdx.x * blockDim.x + threadIdx.x;
  if (i >= (long)NN * HH) return;
  int n = (int)(i >> 7), c = (int)(i & 127);
  float s = 0.f;
#pragma unroll
  for (int f = 0; f < 9; ++f) s += emb[((long)(f * 64 + x[n * 9 + f])) * HH + c];
  h[i] = s;
}

__global__ void k_bond(const int* __restrict__ idx, const float* __restrict__ emb,
                       float* __restrict__ e) {
  long i = (long)blockIdx.x * blockDim.x + threadIdx.x;
  if (i >= (long)EE * HH) return;
  int n = (int)(i >> 7), c = (int)(i & 127);
  float s = 0.f;
#pragma unroll
  for (int f = 0; f < 3; ++f) s += emb[((long)(f * 8 + idx[n * 3 + f])) * HH + c];
  e[i] = s;
}

// ---------------- fused edge/message kernel ----------------
// 64 edges per block (8 waves).  ea = e@edge_W+b (WMMA) into LDS, then
// m = [h_dst|h_src|ea] @ pre_W + pre_b (WMMA, K=384, 256 cols).  Each wave
// owns a col-tile and carries 4 row-tile accumulators so every B fragment
// feeds 4 WMMAs.  Outputs scattered straight into segment sum/sq/min/max.
#define APAD 388   // 384 padded (stride%64==4 -> conflict free)
#define EPAD 132   // 128 padded
__global__ void k_msg(const float* __restrict__ h, const float* __restrict__ e,
                      const int* __restrict__ ei,
                      const float* __restrict__ edgeW, const float* __restrict__ edgeB,
                      const float* __restrict__ preW, const float* __restrict__ preB,
                      float* __restrict__ s1, float* __restrict__ s2,
                      unsigned* __restrict__ mnb, unsigned* __restrict__ mxb, int l) {
  extern __shared__ float sm[];
  float* A  = sm;               // 64*APAD  ([h_dst|h_src|ea] rows)
  float* eL = A + 64 * APAD;    // 64*EPAD
  __shared__ int dstl[64], srcl[64];
  const int tid = threadIdx.x;  // 256 threads = 8 waves
  const int base = blockIdx.x * 64;
  if (tid < 64) {
    srcl[tid] = ei[base + tid];
    dstl[tid] = ei[EE + base + tid];
  }
  __syncthreads();
  // async stage: e rows (16x128) and gathered h rows (16x256), 16B chunks
#pragma unroll
  for (int i = 0; i < 2; ++i) {
    int c = tid + i * 256;
    int r = c >> 5, cj = (c & 31) << 2;
    async_b128(&eL[r * EPAD + cj], &e[(long)(base + r) * HH + cj]);
  }
#pragma unroll
  for (int i = 0; i < 4; ++i) {
    int c = tid + i * 256;
    int r = c >> 6, cj = (c & 63) << 2;
    const float* src = (cj < 128) ? &h[(long)dstl[r] * HH + cj]
                                  : &h[(long)srcl[r] * HH + (cj - 128)];
    async_b128(&A[r * APAD + cj], src);
  }
  async_wait0();
  __syncthreads();
  const int lane = tid & 31, w = tid >> 5;
  const int rowm = lane & 15, ksel = (lane >> 4) << 1;
  // ---- ea GEMM: wave w -> col tile w; 4 row tiles share each B fragment
  {
    const int j = w * 16 + rowm;
    const float* B = edgeW + (long)l * HH * HH;
    v8f acc[4] = {};
    for (int k = 0; k < 128; k += 4) {
      int kb = k + ksel;
      v2f bv;
      bv.x = B[kb * HH + j];
      bv.y = B[(kb + 1) * HH + j];
#pragma unroll
      for (int rt = 0; rt < 4; ++rt) {
        v2f a;
        int ra = (rt * 16 + rowm) * EPAD;
        a.x = eL[ra + kb];
        a.y = eL[ra + kb + 1];
        acc[rt] = wmma4(a, bv, acc[rt]);
      }
    }
    const float bias = edgeB[l * HH + j];
#pragma unroll
    for (int rt = 0; rt < 4; ++rt)
#pragma unroll
      for (int v = 0; v < 8; ++v) {
        int rr = rt * 16 + ((lane < 16) ? v : v + 8);
        A[rr * APAD + 256 + j] = acc[rt][v] + bias;
      }
  }
  __syncthreads();
  // ---- pre GEMM: 16 col tiles, 2 per wave; 4 row tiles share B
  for (int cti = 0; cti < 2; ++cti) {
    const int ct = w * 2 + cti;
    const int j = ct * 16 + rowm;   // global col 0..255 = t*128+jj
    const int t = j >> 7, jj = j & 127;
    const float* B = preW + (long)(l * 2 + t) * 384 * HH;
    v8f acc[4] = {};
    for (int k = 0; k < 384; k += 4) {
      int kb = k + ksel;
      if ((k & 127) == 0 && k + 128 < 384)
        __builtin_prefetch(&B[(k + 128) * HH + jj], 0, 1);
      v2f bv;
      bv.x = B[kb * HH + jj];
      bv.y = B[(kb + 1) * HH + jj];
#pragma unroll
      for (int rt = 0; rt < 4; ++rt) {
        v2f a;
        int ra = (rt * 16 + rowm) * APAD;
        a.x = A[ra + kb];
        a.y = A[ra + kb + 1];
        acc[rt] = wmma4(a, bv, acc[rt]);
      }
    }
    const float bias = preB[(l * 2 + t) * HH + jj];
#pragma unroll
    for (int rt = 0; rt < 4; ++rt)
#pragma unroll
      for (int v = 0; v < 8; ++v) {
        int er = rt * 16 + ((lane < 16) ? v : v + 8);
        float val = acc[rt][v] + bias;
        int d = dstl[er];
        long off = (long)d * 256 + j;
        unsigned enc = fenc(val);
        atomicAdd(&s1[off], val);
        atomicAdd(&s2[off], val * val);
        atomicMin(&mnb[off], enc);
        atomicMax(&mxb[off], enc);
      }
  }
}

// ---------------- fused node update kernel ----------------
// 32 nodes per block (8 waves).  Build agg (mean|min|max|std per tower) in
// LDS, post-GEMM (K=1664, A generated on the fly with amp/att scaling) then
// lin-GEMM.  2 row tiles per wave share B fragments.
#define HPAD 132
#define GPAD 1028
__device__ __forceinline__ float getA(const float* hl, const float* ag,
                                      const float* ampl, const float* attl,
                                      int r, int k, int t) {
  if (k < 128) return hl[r * HPAD + k];
  int kk = k - 128;
  int b3 = kk >> 9;        // 0: agg, 1: agg*amp, 2: agg*att
  int inner = kk & 511;    // blk*128+ch; blk in {mean,min,max,std}
  float base = ag[r * GPAD + t * 512 + inner];
  if (b3 == 0) return base;
  return (b3 == 1) ? base * ampl[r] : base * attl[r];
}

__global__ void k_post(const float* __restrict__ h, const float* __restrict__ s1,
                       const float* __restrict__ s2, const unsigned* __restrict__ mnb,
                       const unsigned* __restrict__ mxb, const int* __restrict__ cnt,
                       const float* __restrict__ postW, const float* __restrict__ postB,
                       const float* __restrict__ linW, const float* __restrict__ linB,
                       float* __restrict__ o, int l) {
  extern __shared__ float sm[];
  float* hl = sm;                   // 32*HPAD
  float* ag = hl + 32 * HPAD;       // 32*GPAD
  float* ol = ag + 32 * GPAD;       // 32*HPAD
  float* ampl = ol + 32 * HPAD;     // 32
  float* attl = ampl + 32;          // 32
  float* degl = attl + 32;          // 32
  float* hasl = degl + 32;          // 32
  const int tid = threadIdx.x;      // 256 threads = 8 waves
  const int nb = blockIdx.x * 32;
  if (tid < 32) {
    int node = nb + tid;
    float c = (node < NN) ? (float)cnt[node] : 0.f;
    float d = fmaxf(c, 1.f);
    float ld = logf(d + 1.f);
    degl[tid] = d;
    ampl[tid] = ld / AVGDEG;
    attl[tid] = AVGDEG / ld;
    hasl[tid] = (c > 0.f) ? 1.f : 0.f;
  }
  // async stage h rows (zeros for OOB rows)
#pragma unroll
  for (int i = 0; i < 4; ++i) {
    int c = tid + i * 256;
    int r = c >> 5, cj = (c & 31) << 2;
    int node = nb + r;
    if (node < NN) {
      async_b128(&hl[r * HPAD + cj], &h[(long)node * HH + cj]);
    } else {
      hl[r * HPAD + cj] = 0.f;
      hl[r * HPAD + cj + 1] = 0.f;
      hl[r * HPAD + cj + 2] = 0.f;
      hl[r * HPAD + cj + 3] = 0.f;
    }
  }
  __syncthreads();  // degl/hasl ready for agg build
  for (int i = tid; i < 32 * 1024; i += 256) {
    int r = i >> 10, q = i & 1023;
    int node = nb + r;
    float val = 0.f;
    if (node < NN) {
      int t = q >> 9, blk = (q >> 7) & 3, ch = q & 127;
      long off = (long)node * 256 + t * 128 + ch;
      float d = degl[r];
      if (blk == 0) val = s1[off] / d;
      else if (blk == 1) val = hasl[r] * fdec(mnb[off]);
      else if (blk == 2) val = hasl[r] * fdec(mxb[off]);
      else {
        float me = s1[off] / d;
        float v2 = s2[off] / d - me * me;
        val = sqrtf(fmaxf(v2, 0.f) + EPS);
      }
    }
    ag[r * GPAD + q] = val;
  }
  async_wait0();
  __syncthreads();
  const int lane = tid & 31, w = tid >> 5;
  const int rowm = lane & 15, ksel = (lane >> 4) << 1;
  // ---- post GEMM: wave -> (tower t=w>>2, col tile ct=w&3); 2 row tiles
  {
    const int t = w >> 2, ct = w & 3;
    const int j = ct * 16 + rowm;   // 0..63
    const float* B = postW + (long)(l * 2 + t) * 1664 * 64;
    v8f acc[2] = {};
    for (int k = 0; k < 1664; k += 4) {
      int kb = k + ksel;
      if ((k & 255) == 0 && k + 256 < 1664)
        __builtin_prefetch(&B[(long)(k + 256) * 64 + j], 0, 1);
      v2f bv;
      bv.x = B[(long)kb * 64 + j];
      bv.y = B[(long)(kb + 1) * 64 + j];
#pragma unroll
      for (int rt = 0; rt < 2; ++rt) {
        v2f a;
        a.x = getA(hl, ag, ampl, attl, rt * 16 + rowm, kb, t);
        a.y = getA(hl, ag, ampl, attl, rt * 16 + rowm, kb + 1, t);
        acc[rt] = wmma4(a, bv, acc[rt]);
      }
    }
    const float bias = postB[(l * 2 + t) * 64 + j];
#pragma unroll
    for (int rt = 0; rt < 2; ++rt)
#pragma unroll
      for (int v = 0; v < 8; ++v) {
        int rr = rt * 16 + ((lane < 16) ? v : v + 8);
        ol[rr * HPAD + t * 64 + j] = acc[rt][v] + bias;  // o.reshape(N,T*FOUT)
      }
  }
  __syncthreads();
  // ---- lin GEMM: 128x128, wave w -> col tile w; 2 row tiles
  {
    const int j = w * 16 + rowm;
    const float* B = linW + (long)l * HH * HH;
    v8f acc[2] = {};
    for (int k = 0; k < 128; k += 4) {
      int kb = k + ksel;
      v2f bv;
      bv.x = B[kb * HH + j];
      bv.y = B[(kb + 1) * HH + j];
#pragma unroll
      for (int rt = 0; rt < 2; ++rt) {
        v2f a;
        int ra = (rt * 16 + rowm) * HPAD;
        a.x = ol[ra + kb];
        a.y = ol[ra + kb + 1];
        acc[rt] = wmma4(a, bv, acc[rt]);
      }
    }
    const float bias = linB[l * HH + j];
#pragma unroll
    for (int rt = 0; rt < 2; ++rt)
#pragma unroll
      for (int v = 0; v < 8; ++v) {
        int rr = rt * 16 + ((lane < 16) ? v : v + 8);
        int node = nb + rr;
        if (node < NN) o[(long)node * HH + j] = acc[rt][v] + bias;
      }
  }
}

// ---------------- batch norm (training mode over M rows) ----------------
__global__ void k_bn_reduce(const float* __restrict__ x, float* s, float* q,
                            int M, int C) {
  int c = threadIdx.x;  // blockDim == C
  float a = 0.f, b = 0.f;
  for (int r = blockIdx.x; r < M; r += gridDim.x) {
    float v = x[(long)r * C + c];
    a += v;
    b += v * v;
  }
  atomicAdd(&s[c], a);
  atomicAdd(&q[c], b);
}

__global__ void k_bn_apply(const float* __restrict__ x, float* __restrict__ y,
                           const float* __restrict__ s, const float* __restrict__ q,
                           const float* __restrict__ g, const float* __restrict__ b,
                           int M, int C, int relu) {
  long i = (long)blockIdx.x * blockDim.x + threadIdx.x;
  if (i >= (long)M * C) return;
  int c = (int)(i % C);
  float mu = s[c] / (float)M;
  float var = q[c] / (float)M - mu * mu;
  float sc = rsqrtf(var + EPS) * g[c];
  float v = (x[i] - mu) * sc + b[c];
  if (relu) v = fmaxf(v, 0.f);
  y[i] = v;
}

// ---------------- pooling ----------------
__global__ void k_pool(const float* __restrict__ h, const int* __restrict__ batch,
                       float* __restrict__ g) {
  long i = (long)blockIdx.x * blockDim.x + threadIdx.x;
  if (i >= (long)NN * HH) return;
  int n = (int)(i >> 7), c = (int)(i & 127);
  atomicAdd(&g[(long)batch[n] * HH + c], h[i]);
}

// ---------------- generic WMMA GEMM (small MLP/head) ----------------
__global__ void k_gemm_bias(const float* __restrict__ A, const float* __restrict__ B,
                            const float* __restrict__ bias, float* __restrict__ C,
                            int M, int K, int Ncol) {
  const int tilesN = (Ncol + 15) / 16;
  const int tilesM = (M + 15) / 16;
  const int tile = blockIdx.x * (blockDim.x >> 5) + (threadIdx.x >> 5);
  if (tile >= tilesM * tilesN) return;
  const int tm = tile / tilesN, tn = tile % tilesN;
  const int lane = threadIdx.x & 31;
  const int rowm = lane & 15, ksel = (lane >> 4) << 1;
  const int arow = tm * 16 + rowm;
  const int col = tn * 16 + rowm;
  const bool aok = arow < M, cok = col < Ncol;
  v8f c = {};
  for (int k = 0; k < K; k += 4) {
    int kb = k + ksel;
    v2f a, bv;
    a.x = aok ? A[(long)arow * K + kb] : 0.f;
    a.y = aok ? A[(long)arow * K + kb + 1] : 0.f;
    bv.x = cok ? B[(long)kb * Ncol + col] : 0.f;
    bv.y = cok ? B[(long)(kb + 1) * Ncol + col] : 0.f;
    c = wmma4(a, bv, c);
  }
  float bb = cok ? bias[col] : 0.f;
#pragma unroll
  for (int v = 0; v < 8; ++v) {
    int r = tm * 16 + ((lane < 16) ? v : v + 8);
    if (r < M && cok) C[(long)r * Ncol + col] = c[v] + bb;
  }
}

// ---------------- host ----------------
extern "C" void kernel_launch(void* const* d_in, const int* in_sizes, int n_in,
                              void* d_out, int out_size, void* d_ws, size_t ws_size,
                              hipStream_t stream) {
  const int* x = (const int*)d_in[0];
  const int* ei = (const int*)d_in[1];
  const int* batch = (const int*)d_in[2];
  const int* eai = (const int*)d_in[3];
  const float* atom_emb = (const float*)d_in[4];
  const float* bond_emb = (const float*)d_in[5];
  const float* edgeW = (const float*)d_in[6];
  const float* edgeB = (const float*)d_in[7];
  const float* preW = (const float*)d_in[8];
  const float* preB = (const float*)d_in[9];
  const float* postW = (const float*)d_in[10];
  const float* postB = (const float*)d_in[11];
  const float* linW = (const float*)d_in[12];
  const float* linB = (const float*)d_in[13];
  const float* bng = (const float*)d_in[14];
  const float* bnb = (const float*)d_in[15];
  const float* mlpW = (const float*)d_in[16];
  const float* mlpB = (const float*)d_in[17];
  const float* mlpG = (const float*)d_in[18];
  const float* mlpBe = (const float*)d_in[19];
  const float* hW1 = (const float*)d_in[20];
  const float* hb1 = (const float*)d_in[21];
  const float* hg1 = (const float*)d_in[22];
  const float* hbe1 = (const float*)d_in[23];
  const float* hW2 = (const float*)d_in[24];
  const float* hb2 = (const float*)d_in[25];
  const float* hg2 = (const float*)d_in[26];
  const float* hbe2 = (const float*)d_in[27];
  const float* hW3 = (const float*)d_in[28];
  const float* hb3 = (const float*)d_in[29];
  float* out = (float*)d_out;

  char* wsb = (char*)d_ws;
  size_t off = 0;
  auto carve = [&](size_t bytes) -> char* {
    char* p = wsb + off;
    off = (off + bytes + 255) & ~(size_t)255;
    return p;
  };
  float* h = (float*)carve((size_t)NN * HH * 4);
  float* e = (float*)carve((size_t)EE * HH * 4);
  float* s1 = (float*)carve((size_t)NN * 256 * 4);
  float* s2 = (float*)carve((size_t)NN * 256 * 4);
  unsigned* mnb = (unsigned*)carve((size_t)NN * 256 * 4);
  unsigned* mxb = (unsigned*)carve((size_t)NN * 256 * 4);
  float* o = (float*)carve((size_t)NN * HH * 4);
  int* cnt = (int*)carve((size_t)NN * 4);
  float* bns = (float*)carve(128 * 4);
  float* bnq = (float*)carve(128 * 4);
  float* g0 = (float*)carve((size_t)GG * HH * 4);
  float* g1 = (float*)carve((size_t)GG * HH * 4);

  auto fillu = [&](void* p, unsigned v, long n) {
    k_fill<<<(unsigned)((n + 255) / 256), 256, 0, stream>>>((unsigned*)p, v, n);
  };

  // degree + encoders
  fillu(cnt, 0u, NN);
  k_count<<<(EE + 255) / 256, 256, 0, stream>>>(ei, cnt, EE);
  k_atom<<<(unsigned)(((long)NN * HH + 255) / 256), 256, 0, stream>>>(x, atom_emb, h);
  k_bond<<<(unsigned)(((long)EE * HH + 255) / 256), 256, 0, stream>>>(eai, bond_emb, e);

  const size_t smem_msg = (size_t)(64 * APAD + 64 * EPAD) * sizeof(float);
  const size_t smem_post =
      (size_t)(32 * HPAD + 32 * GPAD + 32 * HPAD + 128) * sizeof(float);

  for (int l = 0; l < 3; ++l) {
    fillu(s1, 0u, (long)NN * 256);
    fillu(s2, 0u, (long)NN * 256);
    fillu(mnb, 0xFFFFFFFFu, (long)NN * 256);  // +inf in sortable encoding
    fillu(mxb, 0x00000000u, (long)NN * 256);  // -inf in sortable encoding
    k_msg<<<EE / 64, 256, smem_msg, stream>>>(h, e, ei, edgeW, edgeB, preW, preB,
                                              s1, s2, mnb, mxb, l);
    k_post<<<(NN + 31) / 32, 256, smem_post, stream>>>(h, s1, s2, mnb, mxb, cnt,
                                                       postW, postB, linW, linB,
                                                       o, l);
    fillu(bns, 0u, 128);
    fillu(bnq, 0u, 128);
    k_bn_reduce<<<128, 128, 0, stream>>>(o, bns, bnq, NN, HH);
    k_bn_apply<<<(unsigned)(((long)NN * HH + 255) / 256), 256, 0, stream>>>(
        o, h, bns, bnq, bng + l * HH, bnb + l * HH, NN, HH, 1);
  }

  // graph pooling
  fillu(g0, 0u, (long)GG * HH);
  k_pool<<<(unsigned)(((long)NN * HH + 255) / 256), 256, 0, stream>>>(h, batch, g0);

  auto gemm = [&](const float* A, const float* B, const float* bias, float* C,
                  int M, int K, int Ncol) {
    int tiles = ((M + 15) / 16) * ((Ncol + 15) / 16);
    k_gemm_bias<<<(tiles + 3) / 4, 128, 0, stream>>>(A, B, bias, C, M, K, Ncol);
  };
  auto bn = [&](const float* xin, float* yout, const float* g, const float* b,
                int M, int C, int relu) {
    fillu(bns, 0u, C);
    fillu(bnq, 0u, C);
    k_bn_reduce<<<64, C, 0, stream>>>(xin, bns, bnq, M, C);
    k_bn_apply<<<(unsigned)(((long)M * C + 255) / 256), 256, 0, stream>>>(
        xin, yout, bns, bnq, g, b, M, C, relu);
  };

  // MLP: two 128->128 BN+relu layers
  for (int i = 0; i < 2; ++i) {
    gemm(g0, mlpW + (size_t)i * HH * HH, mlpB + i * HH, g1, GG, HH, HH);
    bn(g1, g0, mlpG + i * HH, mlpBe + i * HH, GG, HH, 1);
  }
  // head: 128->64, 64->32 (BN+relu), 32->3
  gemm(g0, hW1, hb1, g1, GG, 128, 64);
  bn(g1, g0, hg1, hbe1, GG, 64, 1);
  gemm(g0, hW2, hb2, g1, GG, 64, 32);
  bn(g1, g0, hg2, hbe2, GG, 32, 1);
  gemm(g0, hW3, hb3, out, GG, 32, 3);
}